// focal_loss_30863634989162
// MI455X (gfx1250) — compile-verified
//
#include <hip/hip_runtime.h>

// COEF = 0.1 * (1.0 - 0.8)^2 = 0.004
#define COEF 0.004f

typedef __attribute__((ext_vector_type(2))) float v2f;
typedef __attribute__((ext_vector_type(4))) float v4f;
typedef __attribute__((ext_vector_type(4))) int   v4i;
typedef __attribute__((ext_vector_type(8))) float v8f;

// Full-wave (32 lane) f32 sum using the matrix pipe:
// A (16x4 f32): lane L contributes its value at [M = L%16][K = 2*(L/16)],
// second K-slot zeroed. B = all-ones (4x16). Then
//   D[m][n] = sum_k A[m][k] = v[m] + v[m+16]   (independent of n)
// C/D layout: lane L, vgpr r holds D[r + 8*(L/16)][L%16], so summing the 8
// accumulator regs gives lanes 0-15 sum(pair[0..7]) and lanes 16-31
// sum(pair[8..15]); one xor-16 shuffle completes the reduction.
// Must be called with EXEC all-1s (no divergence).
__device__ __forceinline__ float wave_reduce_wmma(float v) {
    v2f a;
    a[0] = v;
    a[1] = 0.0f;
    v2f b;
    b[0] = 1.0f;
    b[1] = 1.0f;
    v8f c = {};
    c = __builtin_amdgcn_wmma_f32_16x16x4_f32(
        /*neg_a=*/false, a, /*neg_b=*/false, b,
        /*c_mod=*/(short)0, c, /*reuse_a=*/false, /*reuse_b=*/false);
    float s = c[0] + c[1] + c[2] + c[3] + c[4] + c[5] + c[6] + c[7];
    s += __shfl_xor(s, 16, 32);
    return s;  // full 32-lane sum, replicated in every lane
}

__global__ void __launch_bounds__(256)
focal_sum_kernel(const float* __restrict__ p,
                 const int* __restrict__ index,
                 float* __restrict__ out,
                 int n) {
    const int nvec   = n >> 2;
    const int tid    = blockIdx.x * blockDim.x + threadIdx.x;
    const int stride = gridDim.x * blockDim.x;

    const v4f* __restrict__ p4 = (const v4f*)p;
    const v4i* __restrict__ i4 = (const v4i*)index;

    float acc = 0.0f;

    // Streaming, coalesced b128 loads with non-temporal hint (read-once data).
    for (int i = tid; i < nvec; i += stride) {
        v4f pv = __builtin_nontemporal_load(p4 + i);
        v4i iv = __builtin_nontemporal_load(i4 + i);
#pragma unroll
        for (int k = 0; k < 4; ++k) {
            float pp = pv[k];
            int   id = iv[k];
            // idx==0 -> log(p); idx==1 -> log(1-p); else log(1)=0.
            float arg = (id == 0) ? pp : ((id == 1) ? (1.0f - pp) : 1.0f);
            acc += __logf(arg);
        }
    }

    // Scalar tail (none for N = 2^24, kept for generality).
    if (tid == 0) {
        for (int j = nvec << 2; j < n; ++j) {
            float pp = p[j];
            int   id = index[j];
            float arg = (id == 0) ? pp : ((id == 1) ? (1.0f - pp) : 1.0f);
            acc += __logf(arg);
        }
    }

    // All threads reconverged here: EXEC is all-1s -> WMMA is legal.
    float wsum = wave_reduce_wmma(acc);

    __shared__ float lds_partial[8];  // 256 threads = 8 wave32
    const int lane = threadIdx.x & 31;
    const int wid  = threadIdx.x >> 5;
    if (lane == 0) lds_partial[wid] = wsum;
    __syncthreads();

    if (wid == 0) {
        float t = (lane < 8) ? lds_partial[lane] : 0.0f;
        t += __shfl_xor(t, 4, 32);
        t += __shfl_xor(t, 2, 32);
        t += __shfl_xor(t, 1, 32);
        if (lane == 0) {
            atomicAdd(out, t * COEF);  // global_atomic_add_f32
        }
    }
}

__global__ void zero_out_kernel(float* out) {
    if (threadIdx.x == 0 && blockIdx.x == 0) out[0] = 0.0f;
}

extern "C" void kernel_launch(void* const* d_in, const int* in_sizes, int n_in,
                              void* d_out, int out_size, void* d_ws, size_t ws_size,
                              hipStream_t stream) {
    const float* p     = (const float*)d_in[0];
    const int*   index = (const int*)d_in[1];
    float*       out   = (float*)d_out;
    const int    n     = in_sizes[0];

    zero_out_kernel<<<1, 32, 0, stream>>>(out);

    const int threads = 256;
    const int blocks  = 2048;  // 16384 wave32 across the device; each thread
                               // handles 8 float4+int4 load pairs at N=2^24.
    focal_sum_kernel<<<blocks, threads, 0, stream>>>(p, index, out, n);
}